// GIN_21174188769406
// MI455X (gfx1250) — compile-verified
//
#include <hip/hip_runtime.h>
#include <math.h>

// GIN forward for MI455X (gfx1250, wave32).
// Roofline: 19.7 GFLOP of GEMM vs ~614M L2-resident f32 atomics for the edge
// scatter -> scatter dominates; whole working set (~130MB) fits in 192MB L2.
// GEMMs fused per 128-row stripe with V_WMMA_F32_16X16X4_F32 (exact fp32
// numerics); weights staged into LDS via the Tensor Data Mover (TDM).

#define DIMN 128
constexpr int N_NODES_C = 100000;
constexpr int N_EDGES_C = 1600000;
constexpr int N_LAYERS_C = 3;

typedef __attribute__((ext_vector_type(2))) float v2f;
typedef __attribute__((ext_vector_type(8))) float v8f;

#if __has_builtin(__builtin_amdgcn_tensor_load_to_lds)
#define USE_TDM 1
typedef unsigned int u32x4 __attribute__((ext_vector_type(4)));
typedef int i32x4 __attribute__((ext_vector_type(4)));
typedef int i32x8 __attribute__((ext_vector_type(8)));

// Issue one TDM DMA: 128x128 f32 tile, global (row-major) -> LDS (linear).
// D# layout per CDNA5 ISA (groups 0/1; 2D tile so
// groups 2/3 are zero).
__device__ __forceinline__ void tdm_load_w128(const float* g,
                                              unsigned lds_byte_off) {
  unsigned long long ga = (unsigned long long)(const void*)g;
  u32x4 g0;
  g0[0] = 1u;                                   // count=1 (valid), user mode
  g0[1] = lds_byte_off;                         // lds_addr [63:32]
  g0[2] = (unsigned)ga;                         // global_addr [95:64]
  g0[3] = ((unsigned)(ga >> 32) & 0x01FFFFFFu)  // global_addr [120:96]
          | (2u << 30);                         // type=2 ("image") [127:126]
  i32x8 g1;
  g1[0] = 0x00020000;                 // wg_mask=0, data_size=2 (4 bytes)
  g1[1] = (int)(128u << 16);          // tensor_dim0 = 128   (bits 79:48)
  g1[2] = (int)(128u << 16);          // tensor_dim1 = 128   (bits 111:80)
  g1[3] = (int)(128u << 16);          // tile_dim0   = 128   (bits 127:112)
  g1[4] = 128;                        // tile_dim1=128, tile_dim2=0
  g1[5] = 128;                        // tensor_dim0_stride = 128
  g1[6] = (int)((128u * 128u) << 16); // tensor_dim1_stride = 16384 (low16)
  g1[7] = 0;                          //   (high bits zero)
  i32x4 gz = {0, 0, 0, 0};
#if defined(__clang_major__) && __clang_major__ >= 23
  i32x8 gz8 = {0, 0, 0, 0, 0, 0, 0, 0};
  __builtin_amdgcn_tensor_load_to_lds(g0, g1, gz, gz, gz8, 0);
#else
  __builtin_amdgcn_tensor_load_to_lds(g0, g1, gz, gz, 0);
#endif
}
#else
#define USE_TDM 0
#endif

// ---------------------------------------------------------------- copy h -> agg
__global__ void gin_copy_kernel(const float4* __restrict__ in,
                                float4* __restrict__ out, int n4) {
  int i = blockIdx.x * blockDim.x + threadIdx.x;
  if (i < n4) out[i] = in[i];
}

// ------------------------------------------------- scatter-add: out[dst] += h[src]
// One 32-lane group per edge; each lane handles 4 consecutive dims (float4
// gather, 4x non-returning global_atomic_add_f32).
__global__ void gin_scatter_kernel(const float* __restrict__ h,
                                   const int* __restrict__ src,
                                   const int* __restrict__ dst,
                                   float* __restrict__ out, int nE) {
  int gid = blockIdx.x * blockDim.x + threadIdx.x;
  int e = gid >> 5;
  int lane = gid & 31;
  if (e >= nE) return;
  int s = src[e];
  int d = dst[e];
  const float4 v = *(const float4*)(h + (size_t)s * DIMN + lane * 4);
  float* o = out + (size_t)d * DIMN + lane * 4;
  atomicAdd(o + 0, v.x);
  atomicAdd(o + 1, v.y);
  atomicAdd(o + 2, v.z);
  atomicAdd(o + 3, v.w);
}

// ------------------------------------------- fused  relu(W2 . relu(bn(W1.h+b1)) + b2)
// 256 threads = 8 wave32; each wave owns a 16-row tile; block covers 128 rows.
// W1,W2 loaded by TDM into LDS (64KB each) + 8x8KB per-wave staging.
__global__ __launch_bounds__(256, 1) void gin_mlp_kernel(
    const float* __restrict__ hin,
    const float* __restrict__ W1, const float* __restrict__ b1,
    const float* __restrict__ gamma, const float* __restrict__ beta,
    const float* __restrict__ rmean, const float* __restrict__ rvar,
    const float* __restrict__ W2, const float* __restrict__ b2,
    float* __restrict__ hout, int nRows) {
  __shared__ float W1s[DIMN * DIMN];        // 64 KB
  __shared__ float W2s[DIMN * DIMN];        // 64 KB
  __shared__ float stage[8][16 * DIMN];     // 64 KB (8 KB per wave)
  __shared__ float s1[DIMN], t1[DIMN], b2s[DIMN];

  const int tid = threadIdx.x;
#if USE_TDM
  if (tid < 32) {  // wave 0 issues both DMAs (TDM ignores EXEC, per-wave op)
    tdm_load_w128(W1, (unsigned)(unsigned long long)(void*)&W1s[0]);
    tdm_load_w128(W2, (unsigned)(unsigned long long)(void*)&W2s[0]);
  }
#else
  for (int i = tid; i < DIMN * DIMN; i += 256) {
    W1s[i] = W1[i];
    W2s[i] = W2[i];
  }
#endif
  // Other waves overlap the BN constant prep with the weight DMA.
  if (tid >= 128 && tid < 128 + DIMN) {
    const int c = tid - 128;
    float inv = rsqrtf(rvar[c] + 1e-5f);
    float s = gamma[c] * inv;
    s1[c] = s;                                   // folded BN scale
    t1[c] = (b1[c] - rmean[c]) * s + beta[c];    // folded b1 + BN shift
    b2s[c] = b2[c];
  }
#if USE_TDM
  if (tid < 32) __builtin_amdgcn_s_wait_tensorcnt((short)0);
#endif
  __syncthreads();

  const int wave = tid >> 5;
  const int lane = tid & 31;
  const int lhi = lane >> 4;   // 0 for lanes 0-15, 1 for lanes 16-31
  const int l15 = lane & 15;
  float* st = &stage[wave][0];

  const int row0 = blockIdx.x * 128 + wave * 16;
  int arow = row0 + l15;                 // A-fragment row for this lane
  if (arow >= nRows) arow = nRows - 1;   // branchless clamp: EXEC stays all-1s
  const float* aptr = hin + (size_t)arow * DIMN + 2 * lhi;

  // ---------------- GEMM1: acc = h_tile(16x128) @ W1(128x128) ----------------
  v8f acc[8];
#pragma unroll
  for (int nt = 0; nt < 8; ++nt)
#pragma unroll
    for (int r = 0; r < 8; ++r) acc[nt][r] = 0.f;

#pragma unroll 4
  for (int kk = 0; kk < DIMN; kk += 4) {
    // A 16x4 layout: lane l holds M=l&15, K = kk + v + 2*(l>>4)
    const float2 av = *(const float2*)(aptr + kk);
    v2f a;
    a[0] = av.x;
    a[1] = av.y;
    const int krow = kk + 2 * lhi;
#pragma unroll
    for (int nt = 0; nt < 8; ++nt) {
      // B 4x16 layout: lane l holds N = nt*16 + (l&15), K = kk + v + 2*(l>>4)
      const int n = nt * 16 + l15;
      v2f b;
      b[0] = W1s[krow * DIMN + n];
      b[1] = W1s[(krow + 1) * DIMN + n];
      acc[nt] = __builtin_amdgcn_wmma_f32_16x16x4_f32(
          false, a, false, b, (short)0, acc[nt], false, false);
    }
  }

  // BN + ReLU epilogue; restage C layout -> row-major 16x128 in LDS
#pragma unroll
  for (int nt = 0; nt < 8; ++nt) {
    const int n = nt * 16 + l15;
    const float s = s1[n];
    const float t = t1[n];
#pragma unroll
    for (int r = 0; r < 8; ++r) {
      // C layout: element (M = r + 8*lhi, N = n)
      float v = acc[nt][r] * s + t;
      v = fmaxf(v, 0.f);
      st[(r + 8 * lhi) * DIMN + n] = v;
    }
  }
  __syncthreads();

  // ---------------- GEMM2: acc2 = relu1_tile(16x128) @ W2(128x128) ----------------
  v8f acc2[8];
#pragma unroll
  for (int nt = 0; nt < 8; ++nt)
#pragma unroll
    for (int r = 0; r < 8; ++r) acc2[nt][r] = 0.f;

  const float* a2ptr = st + l15 * DIMN + 2 * lhi;
#pragma unroll 4
  for (int kk = 0; kk < DIMN; kk += 4) {
    const float2 av = *(const float2*)(a2ptr + kk);  // ds_load_b64
    v2f a;
    a[0] = av.x;
    a[1] = av.y;
    const int krow = kk + 2 * lhi;
#pragma unroll
    for (int nt = 0; nt < 8; ++nt) {
      const int n = nt * 16 + l15;
      v2f b;
      b[0] = W2s[krow * DIMN + n];
      b[1] = W2s[(krow + 1) * DIMN + n];
      acc2[nt] = __builtin_amdgcn_wmma_f32_16x16x4_f32(
          false, a, false, b, (short)0, acc2[nt], false, false);
    }
  }

  // bias + ReLU + store (guard tail rows only at the store)
#pragma unroll
  for (int nt = 0; nt < 8; ++nt) {
    const int n = nt * 16 + l15;
    const float bb = b2s[n];
#pragma unroll
    for (int r = 0; r < 8; ++r) {
      float v = acc2[nt][r] + bb;
      v = fmaxf(v, 0.f);
      const int row = row0 + r + 8 * lhi;
      if (row < nRows) hout[(size_t)row * DIMN + n] = v;
    }
  }
}

// ---------------------------------------------------------------- log_softmax
// One wave32 per row; each lane owns 4 contiguous columns.
__global__ void gin_logsoftmax_kernel(const float* __restrict__ h,
                                      float* __restrict__ out, int nRows) {
  int gid = blockIdx.x * blockDim.x + threadIdx.x;
  int row = gid >> 5;
  int lane = threadIdx.x & 31;
  if (row >= nRows) return;
  const float4 v = *(const float4*)(h + (size_t)row * DIMN + lane * 4);
  float m = fmaxf(fmaxf(v.x, v.y), fmaxf(v.z, v.w));
#pragma unroll
  for (int o = 16; o > 0; o >>= 1) m = fmaxf(m, __shfl_xor(m, o, 32));
  float s = expf(v.x - m) + expf(v.y - m) + expf(v.z - m) + expf(v.w - m);
#pragma unroll
  for (int o = 16; o > 0; o >>= 1) s += __shfl_xor(s, o, 32);
  const float lg = m + logf(s);
  float4 r;
  r.x = v.x - lg;
  r.y = v.y - lg;
  r.z = v.z - lg;
  r.w = v.w - lg;
  *(float4*)(out + (size_t)row * DIMN + lane * 4) = r;
}

// ---------------------------------------------------------------- launch
extern "C" void kernel_launch(void* const* d_in, const int* in_sizes, int n_in,
                              void* d_out, int out_size, void* d_ws,
                              size_t ws_size, hipStream_t stream) {
  const float* x = (const float*)d_in[0];
  const int* ei = (const int*)d_in[1];  // (2, E): row0 = src, row1 = dst
  const float* W1 = (const float*)d_in[2];
  const float* b1 = (const float*)d_in[3];
  const float* gamma = (const float*)d_in[4];
  const float* beta = (const float*)d_in[5];
  const float* rmean = (const float*)d_in[6];
  const float* rvar = (const float*)d_in[7];
  const float* W2 = (const float*)d_in[8];
  const float* b2 = (const float*)d_in[9];

  float* hA = (float*)d_ws;                       // 51.2 MB
  float* hB = hA + (size_t)N_NODES_C * DIMN;      // 51.2 MB

  const int n4 = (N_NODES_C * DIMN) / 4;
  const int copyBlocks = (n4 + 255) / 256;
  const int scatBlocks = (N_EDGES_C * 32 + 255) / 256;
  const int mlpBlocks = (N_NODES_C + 127) / 128;
  const int lsBlocks = (N_NODES_C * 32 + 255) / 256;

  const float* cur = x;
  for (int l = 0; l < N_LAYERS_C; ++l) {
    // hB = cur  (the "(1+eps)*x" term, eps = 0)
    gin_copy_kernel<<<copyBlocks, 256, 0, stream>>>((const float4*)cur,
                                                    (float4*)hB, n4);
    // hB[dst] += cur[src]
    gin_scatter_kernel<<<scatBlocks, 256, 0, stream>>>(
        cur, ei, ei + N_EDGES_C, hB, N_EDGES_C);
    // hA = relu(W2 . relu(bn(W1 . hB + b1)) + b2)
    gin_mlp_kernel<<<mlpBlocks, 256, 0, stream>>>(
        hB, W1 + (size_t)l * DIMN * DIMN, b1 + l * DIMN, gamma + l * DIMN,
        beta + l * DIMN, rmean + l * DIMN, rvar + l * DIMN,
        W2 + (size_t)l * DIMN * DIMN, b2 + l * DIMN, hA, N_NODES_C);
    cur = hA;
  }
  gin_logsoftmax_kernel<<<lsBlocks, 256, 0, stream>>>(cur, (float*)d_out,
                                                      N_NODES_C);
}